// MultiTaskLoss_44770739093769
// MI455X (gfx1250) — compile-verified
//
#include <hip/hip_runtime.h>

#define DT 0.01f
#define EPS 1e-6f

// Fixed problem geometry from the reference setup.
#define BB 64
#define TT 65536
#define CHUNK_CH 2048              // elements per channel handled by one scan block
#define CPR (TT / CHUNK_CH)        // 32 chunks per row
#define NSCAN (BB * CPR)           // 2048 scan blocks
#define NE 2048                    // elementwise blocks (same geometry as scan chunks)
#define NTOT (BB * TT)             // 4,194,304 (b,t) pairs

typedef float v2f __attribute__((ext_vector_type(2)));
typedef float v8f __attribute__((ext_vector_type(8)));

// ---------------------------------------------------------------------------
// Kernel 1 (fused): elementwise velocity MSE + Gaussian-NLL terms AND the
// per-chunk vel sums needed by the scan. Block blk covers elements
// [blk*2048, blk*2048+2048) = batch b = blk>>5, chunk c = blk&31 — identical
// geometry to the scan chunks, so the chunk sums come for free (saves a full
// 33.5 MB pass over vel_pred). Also warms L2 (192 MB) with vel for kernel 3.
// ---------------------------------------------------------------------------
__global__ __launch_bounds__(256) void k_fused_stats(const float* __restrict__ vel,
                                                     const float* __restrict__ cov,
                                                     const float* __restrict__ vgt,
                                                     float* __restrict__ partials,
                                                     float* __restrict__ lve_out,
                                                     float* __restrict__ lc_out) {
  int tid = threadIdx.x, blk = blockIdx.x;
  int b = blk >> 5, c = blk & 31;
  const float2* vel2 = (const float2*)vel;
  const float2* vgt2 = (const float2*)vgt;
  float lve = 0.f, lc = 0.f, sx = 0.f, sy = 0.f;
  size_t base = (size_t)blk * (NTOT / NE);
#pragma unroll
  for (int j = 0; j < (NTOT / NE) / 256; ++j) {
    size_t n = base + tid + j * 256;
    float2 vp = vel2[n], vg = vgt2[n];
    float c0 = cov[3 * n + 0], c1 = cov[3 * n + 1], c2 = cov[3 * n + 2];
    sx += vp.x; sy += vp.y;
    float dx = vg.x - vp.x, dy = vg.y - vp.y;
    lve += dx * dx + dy * dy;
    float sxx = __expf(c0) + EPS;
    float syy = __expf(c1) + EPS;
    float det = sxx * syy - c2 * c2 + EPS;
    float mahal = (syy * dx * dx - 2.f * c2 * dx * dy + sxx * dy * dy) / det;
    lc += mahal + __logf(det);
  }
  __shared__ float r1[256], r2[256], r3[256], r4[256];
  r1[tid] = lve; r2[tid] = lc; r3[tid] = sx; r4[tid] = sy;
  __syncthreads();
  for (int s = 128; s > 0; s >>= 1) {
    if (tid < s) {
      r1[tid] += r1[tid + s]; r2[tid] += r2[tid + s];
      r3[tid] += r3[tid + s]; r4[tid] += r4[tid + s];
    }
    __syncthreads();
  }
  if (tid == 0) {
    lve_out[blk] = r1[0];
    lc_out[blk] = r2[0];
    partials[(2 * b + 0) * CPR + c] = r3[0];
    partials[(2 * b + 1) * CPR + c] = r4[0];
  }
}

// ---------------------------------------------------------------------------
// Kernel 2: exclusive scan of chunk sums per (b,ch) row. 128 rows, trivial.
// ---------------------------------------------------------------------------
__global__ void k_scan_partials(const float* __restrict__ partials,
                                float* __restrict__ offsets) {
  int row = threadIdx.x;
  if (row < 2 * BB) {
    float run = 0.f;
    for (int c = 0; c < CPR; ++c) {
      float v = partials[row * CPR + c];
      offsets[row * CPR + c] = run;
      run += v;
    }
  }
}

// ---------------------------------------------------------------------------
// Kernel 3: WMMA prefix scan + position loss.
// Staging now uses CDNA5 async global->LDS (ASYNCcnt path): each lane copies
// one dword straight into LDS with a per-lane destination, deinterleaving the
// (x,y) channels during the copy — no VGPR round-trip, no ds_store traffic.
// LDS byte offsets are the low 32 bits of the generic pointer to the shared
// array (addrspacecast AS3->flat keeps the AS3 offset in the low bits).
// Per wave: 16x16 tiles; exclusive within-row prefix = X * U_strict,
// row sums = X * 1, both via chained V_WMMA_F32_16X16X4_F32 (exact in f32).
// ---------------------------------------------------------------------------
__global__ __launch_bounds__(256) void k_scan_pos_loss(const float* __restrict__ vel,
                                                       const float* __restrict__ pos_gt,
                                                       const float* __restrict__ offsets,
                                                       float* __restrict__ lvp_out) {
  __shared__ __attribute__((aligned(16))) float Xs[2 * CHUNK_CH];
  __shared__ __attribute__((aligned(16))) float Ps[2 * CHUNK_CH];
  int blk = blockIdx.x;
  int b = blk >> 5, cidx = blk & 31;
  int tid = threadIdx.x;
  size_t base2 = (size_t)b * TT + (size_t)cidx * CHUNK_CH;

  const float2* vel2 = (const float2*)vel;

  // ---- phase 0: async global->LDS staging with channel deinterleave ----
  {
    uint64_t sbase = (uint64_t)(uintptr_t)(vel2 + base2);       // uniform SGPR base
    uint32_t lx = (uint32_t)(uintptr_t)(&Xs[0]);                // LDS offset of ch0
    uint32_t ly = (uint32_t)(uintptr_t)(&Xs[CHUNK_CH]);         // LDS offset of ch1
#pragma unroll
    for (int j = 0; j < CHUNK_CH / 256; ++j) {
      uint32_t i = (uint32_t)(tid + j * 256);
      uint32_t voff = i * 8u;                                   // byte offset of float2
      asm volatile("global_load_async_to_lds_b32 %0, %1, %2 offset:0"
                   :: "v"(lx + 4u * i), "v"(voff), "s"(sbase) : "memory");
      asm volatile("global_load_async_to_lds_b32 %0, %1, %2 offset:4"
                   :: "v"(ly + 4u * i), "v"(voff), "s"(sbase) : "memory");
    }
    asm volatile("s_wait_asynccnt 0x0" ::: "memory");
  }
  __syncthreads();

  // ---- phase 1: per-wave WMMA tile scans (waves 0-3: ch0, 4-7: ch1) ----
  int wave = tid >> 5, lane = tid & 31;
  int half = lane >> 4, li = lane & 15;
  int ch = wave >> 2, wch = wave & 3;

  // B operand (4x16, K x N): VGPR v holds K = v + 2*half, N = li.
  // Strict upper-tri ones: U[k,n] = (k < n).
  v2f Bu[4];
#pragma unroll
  for (int k = 0; k < 4; ++k) {
    int kk = 4 * k + 2 * half;
    Bu[k].x = (kk < li) ? 1.f : 0.f;
    Bu[k].y = (kk + 1 < li) ? 1.f : 0.f;
  }
  v2f Bones; Bones.x = 1.f; Bones.y = 1.f;

  float carry = 0.f;
#pragma unroll
  for (int tile = 0; tile < 2; ++tile) {
    int tb = ch * CHUNK_CH + wch * 512 + tile * 256;
    // A operand (16x4): lane li row, K chunk {4k+2*half, 4k+2*half+1}.
    v2f A[4];
#pragma unroll
    for (int k = 0; k < 4; ++k)
      A[k] = *(const v2f*)&Xs[tb + li * 16 + 4 * k + 2 * half];
    v8f Yx = {0.f, 0.f, 0.f, 0.f, 0.f, 0.f, 0.f, 0.f};  // X * U_strict
    v8f Rm = {0.f, 0.f, 0.f, 0.f, 0.f, 0.f, 0.f, 0.f};  // X * ones -> row sums
#pragma unroll
    for (int k = 0; k < 4; ++k) {
      Yx = __builtin_amdgcn_wmma_f32_16x16x4_f32(false, A[k], false, Bu[k],
                                                 (short)0, Yx, false, false);
      Rm = __builtin_amdgcn_wmma_f32_16x16x4_f32(false, A[k], false, Bones,
                                                 (short)0, Rm, false, false);
    }
    // Rm VGPR v == rowsum[v + 8*half] (uniform across li).
    float s = Rm[0] + Rm[1] + Rm[2] + Rm[3] + Rm[4] + Rm[5] + Rm[6] + Rm[7];
    float cross = __shfl_xor(s, 16, 32);      // other half's 8-row total
    float run = half ? cross : 0.f;           // exclusive row offset
#pragma unroll
    for (int v = 0; v < 8; ++v) {
      // C/D layout: VGPR v, lane -> element (row v+8*half, col li).
      Ps[tb + (v + 8 * half) * 16 + li] = Yx[v] + run + carry;
      run += Rm[v];
    }
    carry += s + cross;                       // tile total
  }
  __syncthreads();

  // Xs is dead now: reuse for wave totals / offsets / reduction scratch.
  float* waveTot = &Xs[0];
  float* waveOff = &Xs[8];
  if (lane == 0) waveTot[wave] = carry;
  __syncthreads();
  if (tid < 2) {  // tid = channel: exclusive scan of 4 wave totals
    float run = 0.f;
    for (int w = 0; w < 4; ++w) { waveOff[tid * 4 + w] = run; run += waveTot[tid * 4 + w]; }
  }
  __syncthreads();

  // ---- phase 2: position loss (pos_pred = DT * (v0 + exclusive prefix)) ----
  const float2* pg2 = (const float2*)pos_gt;
  float v0x = vel[(size_t)b * TT * 2 + 0];
  float v0y = vel[(size_t)b * TT * 2 + 1];
  float off0 = offsets[(2 * b + 0) * CPR + cidx];
  float off1 = offsets[(2 * b + 1) * CPR + cidx];
  float lvp = 0.f;
#pragma unroll
  for (int j = 0; j < CHUNK_CH / 256; ++j) {
    int i = tid + j * 256;
    float2 pg = pg2[base2 + i];
    int w = i >> 9;  // wave (512 elements each) within channel
    float e0 = Ps[i] + waveOff[w] + off0;
    float e1 = Ps[CHUNK_CH + i] + waveOff[4 + w] + off1;
    float d0 = DT * (v0x + e0) - pg.x;
    float d1 = DT * (v0y + e1) - pg.y;
    lvp += d0 * d0 + d1 * d1;
  }
  __syncthreads();
  float* red = &Xs[16];
  red[tid] = lvp;
  __syncthreads();
  for (int s = 128; s > 0; s >>= 1) {
    if (tid < s) red[tid] += red[tid + s];
    __syncthreads();
  }
  if (tid == 0) lvp_out[blk] = red[0];
}

// ---------------------------------------------------------------------------
// Kernel 4: deterministic final reduction + scalar combination.
// ---------------------------------------------------------------------------
__global__ __launch_bounds__(256) void k_finalize(const float* __restrict__ lvp,
                                                  const float* __restrict__ lve,
                                                  const float* __restrict__ lc,
                                                  const float* __restrict__ ldv_p,
                                                  const float* __restrict__ ldc_p,
                                                  float* __restrict__ out) {
  int tid = threadIdx.x;
  float a = 0.f, bsum = 0.f, csum = 0.f;
  for (int i = tid; i < NSCAN; i += 256) a += lvp[i];
  for (int i = tid; i < NE; i += 256) { bsum += lve[i]; csum += lc[i]; }
  __shared__ float r1[256], r2[256], r3[256];
  r1[tid] = a; r2[tid] = bsum; r3[tid] = csum;
  __syncthreads();
  for (int s = 128; s > 0; s >>= 1) {
    if (tid < s) { r1[tid] += r1[tid + s]; r2[tid] += r2[tid + s]; r3[tid] += r3[tid + s]; }
    __syncthreads();
  }
  if (tid == 0) {
    float inv2N = 1.f / (2.f * (float)NTOT);
    float lv_total = (r1[0] + r2[0]) * inv2N;      // mean over (B,T,2) for both terms
    float lc_total = 0.5f * r3[0] / (float)NTOT;   // mean over (B,T)
    float ldv = ldv_p[0], ldc = ldc_p[0];
    float dv = __expf(ldv), dc = __expf(ldc);
    out[0] = lv_total / (2.f * dv * dv) + lc_total / (2.f * dc * dc) + ldv + ldc;
    out[1] = lv_total;
    out[2] = lc_total;
  }
}

// ---------------------------------------------------------------------------
extern "C" void kernel_launch(void* const* d_in, const int* in_sizes, int n_in,
                              void* d_out, int out_size, void* d_ws, size_t ws_size,
                              hipStream_t stream) {
  const float* vel = (const float*)d_in[0];
  const float* cov = (const float*)d_in[1];
  const float* vgt = (const float*)d_in[2];
  const float* pgt = (const float*)d_in[3];
  const float* ldv = (const float*)d_in[4];
  const float* ldc = (const float*)d_in[5];
  float* ws = (float*)d_ws;
  float* partials = ws;             // 4096 floats
  float* offsets  = ws + 4096;      // 4096 floats
  float* lvp      = ws + 8192;      // 2048 floats
  float* lve      = ws + 10240;     // 2048 floats
  float* lc       = ws + 12288;     // 2048 floats
  float* out = (float*)d_out;

  k_fused_stats<<<NE, 256, 0, stream>>>(vel, cov, vgt, partials, lve, lc);
  k_scan_partials<<<1, 128, 0, stream>>>(partials, offsets);
  k_scan_pos_loss<<<NSCAN, 256, 0, stream>>>(vel, pgt, offsets, lvp);
  k_finalize<<<1, 256, 0, stream>>>(lvp, lve, lc, ldv, ldc, out);
}